// SkipGram_53833120088305
// MI455X (gfx1250) — compile-verified
//
#include <hip/hip_runtime.h>
#include <math.h>

// SkipGram fused kernel for MI455X (gfx1250, wave32).
//
// One workgroup (256 thr = 8 waves) per batch element.
//   Phase 1 (all 8 waves): gather-and-mean 21 phrases (1 node from u_table,
//            10 pos + 10 neg contexts from v_table) into LDS with b128 loads;
//            tile pad rows 10..15 zeroed so phase 2 needs no predication.
//            ~88 MB of random row gathers -> this phase is the bandwidth limit.
//   Phase 2 (waves 0,1): two 16x128 @ 128x1 matvecs via 32 chained
//            v_wmma_f32_16x16x4_f32 each (f32 WMMA preserves reference
//            precision; inputs are ~1e-4, dots ~1e-8). All LDS operand loads
//            are unconditional ds_load_b64; the N=0-column mask on B is a
//            float multiply, so EXEC stays all-ones through the WMMA chain.
//   Phase 3 (thread 0): pos-sum, masked exp-sum, log1p, write out[b].

typedef __attribute__((ext_vector_type(2))) float v2f;
typedef __attribute__((ext_vector_type(4))) float f4;
typedef __attribute__((ext_vector_type(8))) float v8f;

namespace {
constexpr int D_DIM = 128;          // embedding dim
constexpr int LMAX  = 8;            // max phrase length
constexpr int W2    = 10;           // positive contexts per batch elem
constexpr int NEGS  = 10;           // negative samples per batch elem
constexpr int PSTR  = D_DIM + 4;    // tile row stride: 132 floats -> 4r+koff
                                    // hits all 64 banks conflict-free
constexpr int TROWS = 16;           // padded tile rows (10 real + 6 zero)
constexpr int NODE_OFF  = 0;                        // node: 128 floats
constexpr int TILEP_OFF = D_DIM;                    // pos tile: 16*132
constexpr int TILEN_OFF = D_DIM + TROWS * PSTR;     // neg tile: 16*132
constexpr int LDS_FLOATS = D_DIM + 2 * TROWS * PSTR;
constexpr int BLOCK = 256;          // 8 wave32 waves
}

__global__ __launch_bounds__(BLOCK)
void skipgram_fused(const float* __restrict__ u_table,
                    const float* __restrict__ v_table,
                    const int*   __restrict__ pos_u_idx,
                    const int*   __restrict__ pos_u_len,
                    const int*   __restrict__ pos_v_idx,
                    const int*   __restrict__ pos_v_len,
                    const int*   __restrict__ neg_v_idx,
                    const int*   __restrict__ neg_v_len,
                    float*       __restrict__ out)
{
    __shared__ float lds[LDS_FLOATS];
    __shared__ float rbuf[32];          // [0..9] pos dots, [16..25] neg dots

    const int b    = blockIdx.x;
    const int tid  = (int)threadIdx.x;
    const int lane = tid & 31;
    const int wave = tid >> 5;

    // ---- Phase 1a: zero the pad rows (10..15) of both tiles ----------------
    // Disjoint from the mean rows, so the single barrier below orders both.
    for (int i = tid; i < 2 * 6 * D_DIM; i += BLOCK) {
        const int t = i / (6 * D_DIM);          // tile 0/1
        const int r = (i / D_DIM) % 6 + 10;     // row 10..15
        const int cidx = i % D_DIM;
        lds[(t ? TILEN_OFF : TILEP_OFF) + r * PSTR + cidx] = 0.f;
    }

    // ---- Phase 1b: phrase means into LDS (all 8 waves, b128 gathers) -------
    for (int ph = wave; ph < 1 + W2 + NEGS; ph += (BLOCK / 32)) {
        const float* table;
        const int*   idxrow;
        int          len, dst;
        if (ph == 0) {
            table = u_table; idxrow = pos_u_idx + (long)b * LMAX;
            len = pos_u_len[b];
            dst = NODE_OFF;                       // node: tight 128-float row
        } else if (ph <= W2) {
            const long n = (long)b * W2 + (ph - 1);
            table = v_table; idxrow = pos_v_idx + n * LMAX;
            len = pos_v_len[n];
            dst = TILEP_OFF + (ph - 1) * PSTR;
        } else {
            const long n = (long)b * NEGS + (ph - 1 - W2);
            table = v_table; idxrow = neg_v_idx + n * LMAX;
            len = neg_v_len[n];
            dst = TILEN_OFF + (ph - 1 - W2) * PSTR;
        }
        f4 acc = {0.f, 0.f, 0.f, 0.f};
        for (int l = 0; l < len; ++l) {
            const long row = idxrow[l];
            const f4* src = (const f4*)(table + row * (long)D_DIM); // b128 load
            acc += src[lane];
        }
        *((f4*)&lds[dst + lane * 4]) = acc / (float)len;            // b128 store
    }
    __syncthreads();

    // ---- Phase 2: dots via chained V_WMMA_F32_16X16X4_F32 ------------------
    // wave 0 -> positive tile, wave 1 -> negative tile.
    // A layout (32-bit A 16x4): lanes 0-15 row M=lane,  K={k0,k0+1};
    //                           lanes 16-31 row M=lane-16, K={k0+2,k0+3}.
    // B layout (4x16): row K striped across lanes; the N=0 column lives in
    //                  lanes with (lane&15)==0 -> arithmetic mask, no EXEC use.
    if (wave < 2) {
        const int   row   = lane & 15;
        const int   kadd  = (lane >> 4) << 1;                 // 0 or 2
        const float nmask = (row == 0) ? 1.0f : 0.0f;
        const float* arow = &lds[(wave ? TILEN_OFF : TILEP_OFF) + row * PSTR + kadd];
        const float* node = &lds[NODE_OFF + kadd];

        v8f c = {};
        #pragma unroll
        for (int k0 = 0; k0 < D_DIM; k0 += 4) {
            const v2f a  = *(const v2f*)(arow + k0);          // ds_load_b64
            const v2f nn = *(const v2f*)(node + k0);          // ds_load_b64 (bcast)
            v2f bb;
            bb.x = nn.x * nmask;
            bb.y = nn.y * nmask;
            // (neg_a, A, neg_b, B, c_mod, C, reuse_a, reuse_b)
            c = __builtin_amdgcn_wmma_f32_16x16x4_f32(
                    false, a, false, bb, (short)0, c, false, false);
        }
        // D column N=0: lane 0 -> M=0..7 in c[0..7]; lane 16 -> M=8..15.
        float* rb = &rbuf[wave * 16];
        if (lane == 0) {
            rb[0] = c[0]; rb[1] = c[1]; rb[2] = c[2]; rb[3] = c[3];
            rb[4] = c[4]; rb[5] = c[5]; rb[6] = c[6]; rb[7] = c[7];
        } else if (lane == 16) {
            rb[8] = c[0]; rb[9] = c[1];
        }
    }
    __syncthreads();

    // ---- Phase 3: final loss ----------------------------------------------
    if (tid == 0) {
        float pos = 0.f;
        #pragma unroll
        for (int k = 0; k < W2; ++k) pos += rbuf[k];
        float neg = 0.f;
        #pragma unroll
        for (int k = 0; k < NEGS; ++k) {
            const float r = rbuf[16 + k];
            neg += (r > 0.f) ? expf(r) : 0.f;
        }
        out[b] = logf(1.f + neg) - pos;     // -(neg_loss + pos_loss)
    }
}

extern "C" void kernel_launch(void* const* d_in, const int* in_sizes, int n_in,
                              void* d_out, int out_size, void* d_ws, size_t ws_size,
                              hipStream_t stream) {
    (void)n_in; (void)d_ws; (void)ws_size; (void)out_size;
    const float* u_table   = (const float*)d_in[0];
    const float* v_table   = (const float*)d_in[1];
    const int*   pos_u_idx = (const int*)d_in[2];
    const int*   pos_u_len = (const int*)d_in[3];
    const int*   pos_v_idx = (const int*)d_in[4];
    const int*   pos_v_len = (const int*)d_in[5];
    const int*   neg_v_idx = (const int*)d_in[6];
    const int*   neg_v_len = (const int*)d_in[7];
    float*       out       = (float*)d_out;

    const int B = in_sizes[3];   // 1024 batch elements (pos_u_len count)
    skipgram_fused<<<B, BLOCK, 0, stream>>>(u_table, v_table,
                                            pos_u_idx, pos_u_len,
                                            pos_v_idx, pos_v_len,
                                            neg_v_idx, neg_v_len,
                                            out);
}